// ModelNew_73315091744385
// MI455X (gfx1250) — compile-verified
//
#include <hip/hip_runtime.h>

typedef __attribute__((ext_vector_type(2))) float v2f;
typedef __attribute__((ext_vector_type(8))) float v8f;
typedef __attribute__((ext_vector_type(4))) int   v4i;

typedef __attribute__((address_space(1))) v4i gv4i;   // global int4
typedef __attribute__((address_space(3))) v4i lv4i;   // LDS int4
typedef __attribute__((address_space(1))) int gi32;   // global int
typedef __attribute__((address_space(3))) int li32;   // LDS int

#define NDIM 4096
#define BT   128   // output block tile (128x128)
#define KT   16    // k tile staged per LDS round
#define ASTR 20    // A LDS row stride  (80B: 16B-aligned for B128 DMA, conflict-free b64 frag loads)
#define BTSTR 20   // Bt LDS col stride (same geometry as A: contiguous b64 frags, conflict-free)

#if __has_builtin(__builtin_amdgcn_global_load_async_to_lds_b128)
#define HAVE_ASYNC128 1
#else
#define HAVE_ASYNC128 0
#endif
#if __has_builtin(__builtin_amdgcn_global_load_async_to_lds_b32)
#define HAVE_ASYNC32 1
#else
#define HAVE_ASYNC32 0
#endif

static __device__ __forceinline__ void async_copy16(const float* src, float* dst) {
#if HAVE_ASYNC128
    __builtin_amdgcn_global_load_async_to_lds_b128(
        (gv4i*)(void*)src, (lv4i*)(void*)dst, 0, 0);
#else
    unsigned lds_addr = (unsigned)(__SIZE_TYPE__)(__attribute__((address_space(3))) void*)(void*)dst;
    unsigned long long gaddr = (unsigned long long)(__SIZE_TYPE__)src;
    asm volatile("global_load_async_to_lds_b128 %0, %1, off"
                 :: "v"(lds_addr), "v"(gaddr) : "memory");
#endif
}

static __device__ __forceinline__ void async_copy4(const float* src, float* dst) {
#if HAVE_ASYNC32
    __builtin_amdgcn_global_load_async_to_lds_b32(
        (gi32*)(void*)src, (li32*)(void*)dst, 0, 0);
#else
    unsigned lds_addr = (unsigned)(__SIZE_TYPE__)(__attribute__((address_space(3))) void*)(void*)dst;
    unsigned long long gaddr = (unsigned long long)(__SIZE_TYPE__)src;
    asm volatile("global_load_async_to_lds_b32 %0, %1, off"
                 :: "v"(lds_addr), "v"(gaddr) : "memory");
#endif
}

static __device__ __forceinline__ void wait_async0() {
#if __has_builtin(__builtin_amdgcn_s_wait_asynccnt)
    __builtin_amdgcn_s_wait_asynccnt(0);
#else
    asm volatile("s_wait_asynccnt 0x0" ::: "memory");
#endif
}

__global__ __launch_bounds__(256, 2)
void trilmm_wmma_f32(const float* __restrict__ A,
                     const float* __restrict__ B,
                     float* __restrict__ C) {
    const int bx = blockIdx.x;   // column block
    const int by = blockIdx.y;   // row block
    const int t  = threadIdx.x;  // 0..255

    const int row0 = by * BT;
    const int col0 = bx * BT;

    if (bx > by) {
        // Strictly upper-triangular block: output is zero (d_out is poisoned).
        const float4 z = make_float4(0.f, 0.f, 0.f, 0.f);
#pragma unroll
        for (int i = 0; i < 16; ++i) {
            int f  = t + i * 256;
            int r  = f >> 5;
            int c4 = f & 31;
            *(float4*)(C + (size_t)(row0 + r) * NDIM + col0 + c4 * 4) = z;
        }
        return;
    }

    __shared__ float Alds[2][BT * ASTR];   // [buf][row][k]  128 x 16 (stride 20)
    __shared__ float Btld[2][BT * BTSTR];  // [buf][col][k]  B transposed, 128 x 16 (stride 20)

    const int w  = t >> 5;    // wave id 0..7 -> tile row within block
    const int ln = t & 31;
    const int hh = ln >> 4;   // half-wave: 0 or 1
    const int li = ln & 15;

    v8f acc[8] = {};

    // ---- panel staging ----
    // Edge k-tiles (diagonal bands of A or B): masked VALU staging (always correct).
    // Interior k-tiles: dense async DMA straight to LDS (B32 copies transpose B on the fly).
    // Selection on scalar values only (kb, bx, by) -> pure scalar branch.
    auto stage = [&](int buf, int kb) {
        const int kbase = kb * KT;
        const bool edge = (kb < bx * 8 + 8) || (kb >= by * 8);
        if (edge) {
            // A panel: 128x16, masked to tril(A)
#pragma unroll
            for (int i = 0; i < 8; ++i) {
                int e  = t + i * 256;
                int rr = e >> 4;
                int kk = e & 15;
                int gr = row0 + rr;
                int gk = kbase + kk;
                float v = A[(size_t)gr * NDIM + gk];
                Alds[buf][rr * ASTR + kk] = (gk <= gr) ? v : 0.0f;
            }
            // B panel: 16x128 -> transposed, masked to tril(B)
#pragma unroll
            for (int i = 0; i < 8; ++i) {
                int e  = t + i * 256;
                int kk = e >> 7;
                int cc = e & 127;
                int gk = kbase + kk;
                int gc = col0 + cc;
                float v = B[(size_t)gk * NDIM + gc];
                Btld[buf][cc * BTSTR + kk] = (gc <= gk) ? v : 0.0f;
            }
        } else {
            // Dense interior tile; no wave divergence: each wave stages 2xB128 (A) + 8xB32 (B).
            // A: wave w covers rows [w*16, w*16+16), 16B (4 floats) per lane.
            {
                const int rq = ln >> 2;        // 0..7 rows per instruction
                const int c4 = (ln & 3) * 4;   // k sub-offset: 0,4,8,12
#pragma unroll
                for (int j = 0; j < 2; ++j) {
                    const int rr = w * 16 + j * 8 + rq;
                    async_copy16(A + (size_t)(row0 + rr) * NDIM + kbase + c4,
                                 &Alds[buf][rr * ASTR + c4]);
                }
            }
            // B: wave w covers k rows {w*2, w*2+1}; each instr reads 32 consecutive
            // cols of one k-row (global-coalesced) and scatters into transposed LDS.
            {
#pragma unroll
                for (int j = 0; j < 8; ++j) {
                    const int kk = w * 2 + (j >> 2);
                    const int cc = (j & 3) * 32 + ln;
                    async_copy4(B + (size_t)(kbase + kk) * NDIM + col0 + cc,
                                &Btld[buf][cc * BTSTR + kk]);
                }
            }
        }
    };

    // ---- WMMA over one staged 16-wide k tile: unconditionally dense ----
    const int arow = (w * 16 + li) * ASTR;
    auto compute = [&](int buf) {
#pragma unroll
        for (int s = 0; s < 4; ++s) {
            const int koff = s * 4 + hh * 2;   // even -> 8B aligned
            v2f a = *(const v2f*)&Alds[buf][arow + koff];
#pragma unroll
            for (int tn = 0; tn < 8; ++tn) {
                v2f b = *(const v2f*)&Btld[buf][(tn * 16 + li) * BTSTR + koff];
                acc[tn] = __builtin_amdgcn_wmma_f32_16x16x4_f32(
                    false, a, false, b, (short)0, acc[tn], false, false);
            }
        }
    };

    // k-tiles where tril(A)*tril(B) can contribute: kb in [bx*8, by*8+7]
    const int kb_beg = bx * 8;
    const int kb_end = by * 8 + 7;

    int p = 0;
    stage(0, kb_beg);  // prologue
    for (int kb = kb_beg; kb <= kb_end; ++kb) {
        wait_async0();       // own async loads for buffer p complete (no-op after VALU staging)
        __syncthreads();     // -> all waves' loads for buffer p visible;
                             //    also proves buffer p^1 is no longer being read
        if (kb < kb_end) stage(p ^ 1, kb + 1);
        compute(p);
        p ^= 1;
    }

    // ---- store with final tril mask ----
    // C/D layout: VGPR v holds M=v (lanes 0-15) / M=v+8 (lanes 16-31), N=lane%16.
#pragma unroll
    for (int tn = 0; tn < 8; ++tn) {
        const int col = col0 + tn * 16 + li;
#pragma unroll
        for (int v = 0; v < 8; ++v) {
            const int row = row0 + w * 16 + hh * 8 + v;
            C[(size_t)row * NDIM + col] = (col <= row) ? acc[tn][v] : 0.0f;
        }
    }
}

extern "C" void kernel_launch(void* const* d_in, const int* in_sizes, int n_in,
                              void* d_out, int out_size, void* d_ws, size_t ws_size,
                              hipStream_t stream) {
    const float* A = (const float*)d_in[0];
    const float* B = (const float*)d_in[1];
    float*       C = (float*)d_out;

    dim3 grid(NDIM / BT, NDIM / BT);  // 32 x 32 blocks
    dim3 block(256);                  // 8 waves (wave32)
    trilmm_wmma_f32<<<grid, block, 0, stream>>>(A, B, C);
}